// EMGDataAugmentation_37529424233298
// MI455X (gfx1250) — compile-verified
//
#include <hip/hip_runtime.h>
#include <stdint.h>

// Problem constants (match reference)
#define B_   512
#define C_   16
#define L_   4000
#define P_   400
#define TPB  256
#define NB128 ((L_ * 4) / 16)        // 1000 x 16B async transfers per channel buffer
#define NGRP  (L_ / 4)               // 1000 groups of 4 consecutive t per channel
#define GPT   4                      // group iterations per thread (ceil(1000/256))

typedef __attribute__((ext_vector_type(4))) float f32x4;

// Low 32 bits of a flat pointer to __shared__ == LDS byte offset within the
// workgroup allocation (flat LDS addr = {shared_aperture, offset[31:0]}).
__device__ __forceinline__ uint32_t lds_off_u32(const void* p) {
  return (uint32_t)(uintptr_t)p;
}

// CDNA5 async DMA: global -> LDS, 16 bytes per enabled lane. Tracked by ASYNCcnt.
__device__ __forceinline__ void async_copy_b128(uint32_t lds_byte_off, const void* gptr) {
  unsigned long long ga = (unsigned long long)(uintptr_t)gptr;
  asm volatile("global_load_async_to_lds_b128 %0, %1, off"
               :: "v"(lds_byte_off), "v"(ga)
               : "memory");
}

__device__ __forceinline__ void wait_async0() {
  asm volatile("s_wait_asynccnt 0" ::: "memory");
}

__global__ __launch_bounds__(TPB)
void emg_aug_kernel(const float* __restrict__ x,
                    const float* __restrict__ jitter,
                    const float* __restrict__ scale_noise,
                    const float* __restrict__ warp_values,
                    float* __restrict__ out)
{
  __shared__ float s_new[P_];    // new_points[b,:]
  __shared__ float s_warp[L_];   // warped[b,:]
  __shared__ float s_x[L_];      // x[b,c,:]        (async-DMA destination)
  __shared__ float s_j[L_];      // jitter[b,c,:]   (async-DMA destination)

  const int   b    = blockIdx.x;
  const int   tid  = threadIdx.x;
  const float step = (float)(L_ - 1) / (float)(P_ - 1);   // orig_points spacing
  const float wstr = 0.1f * (float)L_;                    // warp_strength = 400
  const float scale = scale_noise[b] * 0.1f + 1.0f;       // uniform -> scalar load

  // ---- 1) new_points (endpoints pinned) ----
  for (int j = tid; j < P_; j += TPB) {
    float v = (float)j * step + warp_values[(size_t)b * P_ + j] * wstr;
    if (j == 0)      v = 0.0f;
    if (j == P_ - 1) v = (float)(L_ - 1);
    s_new[j] = v;
  }
  __syncthreads();

  // ---- 2) warped[t] = interp(t, new_points, orig_points) ----
  for (int t = tid; t < L_; t += TPB) {
    const float tf = (float)t;
    // searchsorted(side='right'): first index with xp[idx] > tf
    int lo = 0, hi = P_;
    while (lo < hi) { int mid = (lo + hi) >> 1; if (s_new[mid] <= tf) lo = mid + 1; else hi = mid; }
    int i = lo; if (i < 1) i = 1; if (i > P_ - 1) i = P_ - 1;
    const float xp0 = s_new[i - 1], xp1 = s_new[i];
    const float dx  = xp1 - xp0;
    float f;
    if (dx == 0.0f) f = (float)i * step;                               // fp[i]
    else            f = (float)(i - 1) * step + ((tf - xp0) / dx) * step;
    if (tf < s_new[0])      f = 0.0f;                                  // fp[0]
    if (tf > s_new[P_ - 1]) f = (float)(L_ - 1);                       // fp[-1]
    s_warp[t] = f;
  }
  __syncthreads();

  // ---- 3) per-thread gather descriptors (lo index + weight) for 4 groups of 4
  //         consecutive t; kept in registers and reused for all 16 channels ----
  int   g_lo[GPT * 4];
  float g_w [GPT * 4];
#pragma unroll
  for (int k = 0; k < GPT; ++k) {
    const int g = tid + k * TPB;
#pragma unroll
    for (int e = 0; e < 4; ++e) {
      int   li = 0;
      float w  = 0.0f;
      if (g < NGRP) {
        const int   t  = g * 4 + e;
        const float tf = (float)t;
        int lo = 0, hi = L_;
        while (lo < hi) { int mid = (lo + hi) >> 1; if (s_warp[mid] <= tf) lo = mid + 1; else hi = mid; }
        int i = lo; if (i < 1) i = 1; if (i > L_ - 1) i = L_ - 1;
        const float xp0 = s_warp[i - 1], xp1 = s_warp[i];
        const float dx  = xp1 - xp0;
        li = i - 1;
        w  = (dx == 0.0f) ? 1.0f : (tf - xp0) / dx;
        if (tf < s_warp[0])      { li = 0;      w = 0.0f; }   // -> fp[0]
        if (tf > s_warp[L_ - 1]) { li = L_ - 2; w = 1.0f; }   // -> fp[L-1]
      }
      g_lo[k * 4 + e] = li;
      g_w [k * 4 + e] = w;
    }
  }

  // ---- 4) channel loop: async-stage x/jitter into LDS, fuse augment + warp-lerp,
  //         write 16B non-temporal stores ----
  const uint32_t sx_base = lds_off_u32(&s_x[0]);
  const uint32_t sj_base = lds_off_u32(&s_j[0]);

  for (int c = 0; c < C_; ++c) {
    const float* xc = x      + ((size_t)b * C_ + c) * L_;
    const float* jc = jitter + ((size_t)b * C_ + c) * L_;
#pragma unroll
    for (int k = 0; k < 4; ++k) {
      const int i = tid + k * TPB;
      if (i < NB128) {
        async_copy_b128(sx_base + (uint32_t)i * 16u, xc + (size_t)i * 4);
        async_copy_b128(sj_base + (uint32_t)i * 16u, jc + (size_t)i * 4);
      }
    }
    wait_async0();       // this wave's async DMAs landed in LDS
    __syncthreads();     // every wave's DMAs landed

    f32x4* oc4 = (f32x4*)(out + ((size_t)b * C_ + c) * L_);
#pragma unroll
    for (int k = 0; k < GPT; ++k) {
      const int g = tid + k * TPB;
      if (g < NGRP) {
        f32x4 r;
#pragma unroll
        for (int e = 0; e < 4; ++e) {
          const int   li = g_lo[k * 4 + e];
          const float fa = (s_x[li]     + 0.1f * s_j[li])     * scale;
          const float fb = (s_x[li + 1] + 0.1f * s_j[li + 1]) * scale;
          r[e] = fa + g_w[k * 4 + e] * (fb - fa);
        }
        __builtin_nontemporal_store(r, oc4 + g);   // global_store_b128, NT
      }
    }
    __syncthreads();     // protect s_x/s_j before next channel's DMA overwrites
  }
}

extern "C" void kernel_launch(void* const* d_in, const int* in_sizes, int n_in,
                              void* d_out, int out_size, void* d_ws, size_t ws_size,
                              hipStream_t stream) {
  (void)in_sizes; (void)n_in; (void)out_size; (void)d_ws; (void)ws_size;
  const float* x   = (const float*)d_in[0];   // [B,C,L]
  const float* jit = (const float*)d_in[1];   // [B,C,L]
  const float* sn  = (const float*)d_in[2];   // [B,1,1]
  const float* wv  = (const float*)d_in[3];   // [B,P]
  float* out = (float*)d_out;                 // [B,C,L]
  hipLaunchKernelGGL(emg_aug_kernel, dim3(B_), dim3(TPB), 0, stream,
                     x, jit, sn, wv, out);
}